// PoolLayer_84164179133177
// MI455X (gfx1250) — compile-verified
//
#include <hip/hip_runtime.h>

// ---------------------------------------------------------------------------
// KNN (k=4) graph max-pooling for MI455X (gfx1250, wave32).
//
// dist[b,i,j] = |vi|^2 + |vj|^2 - 2 * <vi, vj>;  per-row argmin is invariant
// to the row-constant |vi|^2, so we rank by  |vj|^2 - 2<vi,vj>  only.
// The <vi,vj> 16x16 tile comes from V_WMMA_F32_16X16X4_F32 (K=3 padded to 4).
// Top-5 selection is a branchless sorted insert on a monotone 64-bit key
// {lo = column index, hi = order-preserving float bits} packed by bit_cast
// (zero-ALU pack) -> v_min_u64/v_max_u64 chains, no EXEC branching.
// All 8 waves of a block cooperate on one 16-row query tile, each sweeping an
// interleaved 1/8 column strip, merging through LDS + the workgroup split
// barrier.  Feature max-pool phase is a coalesced float4 gather/stream
// (memory bound: ~45 MB => ~2us @ 23.3 TB/s).
// ---------------------------------------------------------------------------

typedef float v2f __attribute__((ext_vector_type(2)));
typedef float v8f __attribute__((ext_vector_type(8)));
typedef unsigned long long u64;
typedef unsigned int u32;

#define WAVES_PER_BLOCK 8
#define TILE 16
#define KBEST 5   // 4 neighbors + self
#define NB 4
#define ROWS_PER_WAVE (TILE / WAVES_PER_BLOCK)   // feature-phase rows per wave

// Lexicographic (dist, col) key as a register pair: hi = monotone float bits,
// lo = column.  bit_cast pack => no 64-bit integer arithmetic.
__device__ __forceinline__ u64 make_key(float f, u32 col) {
    u32 u = __float_as_uint(f);
    u ^= ((u32)((int)u >> 31) | 0x80000000u);   // v_ashrrev + v_bitop3
    uint2 p; p.x = col; p.y = u;
    return __builtin_bit_cast(u64, p);
}

// Branchless ascending insert: lowers to v_min_u64 / v_max_u64 pairs.
__device__ __forceinline__ void sorted_insert(u64 bk[KBEST], u64 ck) {
#pragma unroll
    for (int k = 0; k < KBEST; ++k) {
        bool sw = ck < bk[k];
        u64  t  = bk[k];
        bk[k]   = sw ? ck : t;
        ck      = sw ? t  : ck;
    }
}

// ---- Kernel 1: pad vertices to float4 (x, y, z, |v|^2) in workspace -------
__global__ __launch_bounds__(256) void pad_vertices_kernel(
    const float* __restrict__ verts, float4* __restrict__ v4, int total)
{
    int t = blockIdx.x * blockDim.x + threadIdx.x;
    if (t >= total) return;
    float x = verts[3 * t + 0];
    float y = verts[3 * t + 1];
    float z = verts[3 * t + 2];
    v4[t] = make_float4(x, y, z, fmaf(x, x, fmaf(y, y, z * z)));
}

// ---- Kernel 2: WMMA distance tiles + top-5 select + feature max-pool ------
__global__ __launch_bounds__(256) void knn_maxpool_kernel(
    const float4* __restrict__ v4,     // [bs][n] padded vertices
    const float*  __restrict__ feat,   // [bs][n][c]
    const int*    __restrict__ sidx,   // [pool]
    float* __restrict__ out_v,         // [bs][pool][3]
    float* __restrict__ out_f,         // [bs][pool][c]
    int n, int c, int pool)
{
    __shared__ float ldsDist[WAVES_PER_BLOCK][TILE][TILE];
    __shared__ u64   ldsKeys[WAVES_PER_BLOCK][TILE][KBEST];
    __shared__ int   ldsIdx [TILE][NB];

    const int wid  = threadIdx.x >> 5;   // wave id == column strip id
    const int lane = threadIdx.x & 31;
    const int h    = lane >> 4;          // half-wave: 0 or 1
    const int l16  = lane & 15;
    const int b    = blockIdx.y;
    const int rowBase = blockIdx.x * TILE;   // all 8 waves share this tile

    // ---- Load this lane's query (A-matrix operand, clamped => EXEC all 1s)
    const int    slot = rowBase + l16;
    const int    qrow = sidx[min(slot, pool - 1)];
    const float4 q    = v4[(size_t)b * n + qrow];

    // A: 16x4 f32.  vgpr0 = K0 (lanes0-15) | K2 (lanes16-31); vgpr1 = K1 | K3.
    v2f A;
    A.x = h ? q.z : q.x;
    A.y = h ? 0.0f : q.y;

    // Running top-5 as ascending u64 keys.
    u64 bk[KBEST];
#pragma unroll
    for (int k = 0; k < KBEST; ++k) bk[k] = ~0ull;

    // ---- Interleaved column sweep: wave w handles tiles w, w+8, w+16, ... --
    const int nFull = n & ~(TILE - 1);
    for (int colBase = wid * TILE; colBase < nFull;
         colBase += TILE * WAVES_PER_BLOCK) {
        const float4 p = v4[(size_t)b * n + colBase + l16];

        // B: 4x16 f32, K layout mirrored from A.
        v2f B;
        B.x = h ? p.z : p.x;
        B.y = h ? 0.0f : p.y;

        v8f cz = {};
        // D = A x B : D[M=8h+r][N=l16] = <query_M, point_N>
        v8f acc = __builtin_amdgcn_wmma_f32_16x16x4_f32(
            false, A, false, B, (short)0, cz, false, false);

        const float qb = p.w;   // |point|^2 of this lane's column
#pragma unroll
        for (int r = 0; r < 8; ++r)   // rank = |vj|^2 - 2<vi,vj>
            ldsDist[wid][8 * h + r][l16] = fmaf(-2.0f, acc[r], qb);
        __builtin_amdgcn_wave_barrier();

        // Selection: lane handles row l16, columns [colBase+8h, colBase+8h+8).
        const float* rowp = &ldsDist[wid][l16][8 * h];
        float4 va = *(const float4*)(rowp);
        float4 vb = *(const float4*)(rowp + 4);
        float vals[8] = {va.x, va.y, va.z, va.w, vb.x, vb.y, vb.z, vb.w};
        const u32 colLane = (u32)(colBase + 8 * h);
#pragma unroll
        for (int j = 0; j < 8; ++j)
            sorted_insert(bk, make_key(vals[j], colLane + j));
        __builtin_amdgcn_wave_barrier();
    }

    // ---- Tail tile (n not a multiple of 16): wave 0, poisoned OOB keys ----
    if (wid == 0 && nFull < n) {
        const float4 p = v4[(size_t)b * n + min(nFull + l16, n - 1)];
        v2f B;
        B.x = h ? p.z : p.x;
        B.y = h ? 0.0f : p.y;
        v8f cz = {};
        v8f acc = __builtin_amdgcn_wmma_f32_16x16x4_f32(
            false, A, false, B, (short)0, cz, false, false);
        const float qb = p.w;
#pragma unroll
        for (int r = 0; r < 8; ++r)
            ldsDist[0][8 * h + r][l16] = fmaf(-2.0f, acc[r], qb);
        __builtin_amdgcn_wave_barrier();
        const float* rowp = &ldsDist[0][l16][8 * h];
#pragma unroll
        for (int j = 0; j < 8; ++j) {
            int col = nFull + 8 * h + j;
            u64 ck = (col < n) ? make_key(rowp[j], (u32)col) : ~0ull;
            sorted_insert(bk, ck);
        }
    }

    // ---- Intra-wave half merge (wave-synchronous, no block barrier) --------
    if (h) {
#pragma unroll
        for (int k = 0; k < KBEST; ++k) ldsKeys[wid][l16][k] = bk[k];
    }
    __builtin_amdgcn_wave_barrier();
    if (!h) {
        u64 tmp[KBEST];
#pragma unroll
        for (int m = 0; m < KBEST; ++m) tmp[m] = ldsKeys[wid][l16][m];
#pragma unroll
        for (int m = 0; m < KBEST; ++m) sorted_insert(bk, tmp[m]);
#pragma unroll
        for (int k = 0; k < KBEST; ++k) ldsKeys[wid][l16][k] = bk[k];
    }
    __syncthreads();

    // ---- Cross-strip merge: wave 0 folds strips 1..7, publishes neighbors --
    if (wid == 0 && !h) {
        for (int s = 1; s < WAVES_PER_BLOCK; ++s) {
#pragma unroll
            for (int m = 0; m < KBEST; ++m)
                sorted_insert(bk, ldsKeys[s][l16][m]);
        }
        // Drop element 0 (self / nearest), publish 4 neighbor indices.
#pragma unroll
        for (int k = 0; k < NB; ++k) ldsIdx[l16][k] = (int)(u32)bk[k + 1];

        if (slot < pool) {
            float* ov = out_v + ((size_t)b * pool + slot) * 3;
            ov[0] = q.x; ov[1] = q.y; ov[2] = q.z;
        }
    }
    __syncthreads();

    // ---- Feature gather + max-pool: 8 waves x 2 rows, float4 streams ------
    for (int rr = 0; rr < ROWS_PER_WAVE; ++rr) {
        int row = wid * ROWS_PER_WAVE + rr;
        int s2  = rowBase + row;
        if (s2 >= pool) continue;
        int i0 = ldsIdx[row][0];
        int i1 = ldsIdx[row][1];
        int i2 = ldsIdx[row][2];
        int i3 = ldsIdx[row][3];
        const float* f0 = feat + ((size_t)b * n + i0) * c;
        const float* f1 = feat + ((size_t)b * n + i1) * c;
        const float* f2 = feat + ((size_t)b * n + i2) * c;
        const float* f3 = feat + ((size_t)b * n + i3) * c;
        float* op = out_f + ((size_t)b * pool + s2) * c;

        if ((c & 3) == 0) {
            for (int cc = lane * 4; cc < c; cc += 32 * 4) {
                float4 a0 = *(const float4*)(f0 + cc);
                float4 a1 = *(const float4*)(f1 + cc);
                float4 a2 = *(const float4*)(f2 + cc);
                float4 a3 = *(const float4*)(f3 + cc);
                float4 m;
                m.x = fmaxf(fmaxf(a0.x, a1.x), fmaxf(a2.x, a3.x));
                m.y = fmaxf(fmaxf(a0.y, a1.y), fmaxf(a2.y, a3.y));
                m.z = fmaxf(fmaxf(a0.z, a1.z), fmaxf(a2.z, a3.z));
                m.w = fmaxf(fmaxf(a0.w, a1.w), fmaxf(a2.w, a3.w));
                *(float4*)(op + cc) = m;
            }
        } else {
            for (int cc = lane; cc < c; cc += 32) {
                op[cc] = fmaxf(fmaxf(f0[cc], f1[cc]), fmaxf(f2[cc], f3[cc]));
            }
        }
    }
}

// ---------------------------------------------------------------------------
extern "C" void kernel_launch(void* const* d_in, const int* in_sizes, int n_in,
                              void* d_out, int out_size, void* d_ws, size_t ws_size,
                              hipStream_t stream)
{
    const float* vertices = (const float*)d_in[0];
    const float* feat     = (const float*)d_in[1];
    const int*   sidx     = (const int*)d_in[2];

    const int  pool  = in_sizes[2];                       // 1024
    const long bn    = (long)in_sizes[0] / 3;             // bs * n
    const int  c     = (int)((long)in_sizes[1] / bn);     // 256
    const int  bs    = (int)((long)out_size / ((long)pool * (3 + c)));  // 8
    const int  n     = (int)(bn / bs);                    // 4096

    float4* v4    = (float4*)d_ws;                        // bs*n*16 bytes
    float*  out_v = (float*)d_out;
    float*  out_f = out_v + (size_t)bs * pool * 3;

    const int total = bs * n;
    hipLaunchKernelGGL(pad_vertices_kernel,
                       dim3((total + 255) / 256), dim3(256), 0, stream,
                       vertices, v4, total);

    // One 16-row query tile per block; 8 waves split the column sweep.
    dim3 grid((pool + TILE - 1) / TILE, bs);
    hipLaunchKernelGGL(knn_maxpool_kernel, grid, dim3(256), 0, stream,
                       v4, feat, sidx, out_v, out_f, n, c, pool);
}